// Matrix_optimize_87600152969359
// MI455X (gfx1250) — compile-verified
//
#include <hip/hip_runtime.h>
#include <hip/hip_bf16.h>

// out[c] = sum_b ( W[b] / sum_c' |weights[b][c']| ) * |weights[b][c]|
// B = 256 basis rows, C = 131072 classes, all f32.
// Bandwidth-bound (128 MB input, ~1 FLOP/byte). Two passes:
//   pass 1: per-row abs sums  -> scale[b] = W[b]/rowsum[b]   (d_ws)
//   pass 2: weighted column sum via V_WMMA_F32_16X16X4_F32 GEMV.

#define BASIS_NUM 256
#define NUM_CLASSES 131072

typedef __attribute__((ext_vector_type(2))) float v2f;
typedef __attribute__((ext_vector_type(8))) float v8f;

// ---------------- Pass 1: row abs-sum -> scale ----------------
// One block per basis row; 256 threads stream the 512 KB row with b128 loads.
__global__ void __launch_bounds__(256)
rowsum_scale_kernel(const float* __restrict__ W,
                    const float* __restrict__ weights,
                    float* __restrict__ scale) {
    const int b   = blockIdx.x;
    const int tid = threadIdx.x;

    const float4* row = (const float4*)(weights + (size_t)b * NUM_CLASSES);
    float s = 0.0f;
    // 131072 floats = 32768 float4; 128 float4 per thread, fully coalesced.
#pragma unroll 4
    for (int i = tid; i < NUM_CLASSES / 4; i += 256) {
        float4 v = row[i];
        s += fabsf(v.x) + fabsf(v.y) + fabsf(v.z) + fabsf(v.w);
    }

    __shared__ float red[256];
    red[tid] = s;
    __syncthreads();
#pragma unroll
    for (int off = 128; off > 0; off >>= 1) {
        if (tid < off) red[tid] += red[tid + off];
        __syncthreads();
    }
    if (tid == 0) scale[b] = W[b] / red[0];
}

// ---------------- Pass 2: out[c] = sum_b scale[b]*|w[b][c]| ----------------
// GEMV on the WMMA path. Each wave owns 16 output columns (c0..c0+15).
// Per k-step (4 rows contracted):
//   A (16x4, 2 VGPRs): A[m][k] = scale[k0+k], identical for every row m.
//     ISA 32-bit A layout: lanes 0-15 hold K=0 (VGPR0), K=1 (VGPR1);
//     lanes 16-31 hold K=2 (VGPR0), K=3 (VGPR1).
//   B (4x16, 2 VGPRs): B[k][n] = |w[k0+k][c0+n]|, same K half-split, N = lane%16.
// After 64 steps the f32 16x16 accumulator rows are all equal; lanes 0-15
// write acc[0] (M=0, N=lane) to out[c0+lane].
__global__ void __launch_bounds__(256)
gemv_wmma_kernel(const float* __restrict__ weights,
                 const float* __restrict__ scale,
                 float* __restrict__ out) {
    __shared__ float s_scale[BASIS_NUM];
    const int tid = threadIdx.x;
    s_scale[tid] = scale[tid];          // blockDim.x == 256 == BASIS_NUM
    __syncthreads();

    const int lane  = tid & 31;
    const int wave  = tid >> 5;                      // 8 waves per block
    const int c0    = (blockIdx.x * 8 + wave) * 16;  // 16 columns per wave
    const int n     = lane & 15;                     // column within tile
    const int khalf = (lane >> 4) * 2;               // 0 (lanes 0-15) or 2

    const float* colp = weights + (size_t)c0 + n;    // column base for this lane

    v8f acc = {};
#pragma unroll 2
    for (int k0 = 0; k0 < BASIS_NUM; k0 += 4) {
        const int r = k0 + khalf;
        v2f a;
        a.x = s_scale[r];                            // K = khalf     (ds_load_b64)
        a.y = s_scale[r + 1];                        // K = khalf + 1
        v2f bm;
        bm.x = fabsf(colp[(size_t)r * NUM_CLASSES]);         // |w[r  ][c0+n]|
        bm.y = fabsf(colp[(size_t)(r + 1) * NUM_CLASSES]);   // |w[r+1][c0+n]|
        // 8 args: (neg_a, A, neg_b, B, c_mod, C, reuse_a, reuse_b)
        acc = __builtin_amdgcn_wmma_f32_16x16x4_f32(
            /*neg_a=*/false, a, /*neg_b=*/false, bm,
            /*c_mod=*/(short)0, acc, /*reuse_a=*/false, /*reuse_b=*/false);
    }

    if (lane < 16) out[c0 + lane] = acc[0];          // M=0 row of D
}

extern "C" void kernel_launch(void* const* d_in, const int* in_sizes, int n_in,
                              void* d_out, int out_size, void* d_ws, size_t ws_size,
                              hipStream_t stream) {
    const float* W       = (const float*)d_in[0];   // [256]
    const float* weights = (const float*)d_in[1];   // [256, 131072]
    float*       outp    = (float*)d_out;           // [131072]
    float*       scale   = (float*)d_ws;            // 256 floats scratch

    // Pass 1: one block per basis row.
    rowsum_scale_kernel<<<BASIS_NUM, 256, 0, stream>>>(W, weights, scale);

    // Pass 2: 131072 columns / (8 waves * 16 cols) = 1024 blocks.
    gemv_wmma_kernel<<<NUM_CLASSES / 128, 256, 0, stream>>>(weights, scale, outp);
}